// ViSTLoss_45801531245003
// MI455X (gfx1250) — compile-verified
//
#include <hip/hip_runtime.h>
#include <hip/hip_bf16.h>
#include <math.h>

// ---------------------------------------------------------------------------
// Problem constants (from the reference):
//   y_hat: (4, 512, 4000) f32   y: (4, 4000, 512) f32
//   yh = transpose(y_hat) -> (4, 4000, 512)
//   loss = 0.1*sqrt(mean((yh-y)^2)) + 0.5*mean(relu(-yh))
//        + sum over ws in {200,400} of 0.5e-5 * mean_m( dtw_xy - 0.5*(dtw_xx+dtw_yy) )
// ---------------------------------------------------------------------------

#define NB     4
#define T_LEN  4000
#define C_LEN  512
#define TOTAL_ELEMS (4ull * 4000ull * 512ull)
#define GAMMA_INV 10.0f
#define GAMMA     0.1f
#define BIGV      100000000.0f

typedef float v2f __attribute__((ext_vector_type(2)));
typedef float v4f __attribute__((ext_vector_type(4)));
typedef float v8f __attribute__((ext_vector_type(8)));
typedef int   v4i __attribute__((ext_vector_type(4)));

// ---------------------------------------------------------------------------
// Async global->LDS copy (gfx1250 GLOBAL_LOAD_ASYNC_TO_LDS_B128, ASYNCcnt),
// with a synchronous VGPR-bounce fallback if the builtin is unavailable.
// Probe round 3 revealed param0 type: 'int __vector(4) __device__ *' (AS1).
// ---------------------------------------------------------------------------
#if __has_builtin(__builtin_amdgcn_global_load_async_to_lds_b128)
#define HAVE_ASYNC_LDS 1
#else
#define HAVE_ASYNC_LDS 0
#endif

__device__ __forceinline__ void copy16(const float* g, float* l) {
#if HAVE_ASYNC_LDS
    __builtin_amdgcn_global_load_async_to_lds_b128(
        (__attribute__((address_space(1))) v4i*)g,
        (__attribute__((address_space(3))) v4i*)l, 0, 0);
#else
    *(v4f*)l = *(const v4f*)g;
#endif
}

__device__ __forceinline__ void waitAsyncLE4() {
#if HAVE_ASYNC_LDS
#if __has_builtin(__builtin_amdgcn_s_wait_asynccnt)
    __builtin_amdgcn_s_wait_asynccnt(4);
#else
    asm volatile("s_wait_asynccnt 0x4" ::: "memory");
#endif
#endif
}

__device__ __forceinline__ void waitAsync0() {
#if HAVE_ASYNC_LDS
#if __has_builtin(__builtin_amdgcn_s_wait_asynccnt)
    __builtin_amdgcn_s_wait_asynccnt(0);
#else
    asm volatile("s_wait_asynccnt 0x0" ::: "memory");
#endif
#endif
}

// ---------------------------------------------------------------------------
// Kernel 1: tiled transpose y_hat (b,c,t) -> yh (b,t,c)
// ---------------------------------------------------------------------------
__global__ __launch_bounds__(256) void transpose_kernel(const float* __restrict__ yhat,
                                                        float* __restrict__ yh) {
    __shared__ float tile[32][33];
    const int b  = blockIdx.z;
    const int t0 = blockIdx.x * 32;
    const int c0 = blockIdx.y * 32;
    const int tx = threadIdx.x, ty = threadIdx.y;
    const float* src = yhat + (size_t)b * C_LEN * T_LEN;
    float*       dst = yh   + (size_t)b * T_LEN * C_LEN;
#pragma unroll
    for (int r = 0; r < 32; r += 8)
        tile[ty + r][tx] = src[(size_t)(c0 + ty + r) * T_LEN + (t0 + tx)];
    __syncthreads();
#pragma unroll
    for (int r = 0; r < 32; r += 8)
        dst[(size_t)(t0 + ty + r) * C_LEN + (c0 + tx)] = tile[tx][ty + r];
}

// ---------------------------------------------------------------------------
// Kernel 2: row squared-norms for yh and y (wave32: one wave per row)
// ---------------------------------------------------------------------------
__global__ __launch_bounds__(256) void norms_kernel(const float* __restrict__ yh,
                                                    const float* __restrict__ y,
                                                    float* __restrict__ normX,
                                                    float* __restrict__ normY) {
    const int wave = threadIdx.x >> 5;
    const int lane = threadIdx.x & 31;
    const int rid  = blockIdx.x * 8 + wave;
    const float* src;
    float* dst;
    int r;
    if (rid < NB * T_LEN) { src = yh + (size_t)rid * C_LEN; dst = normX; r = rid; }
    else { r = rid - NB * T_LEN; src = y + (size_t)r * C_LEN; dst = normY; }
    float v = 0.0f;
    for (int k = lane; k < C_LEN; k += 32) { float x = src[k]; v += x * x; }
#pragma unroll
    for (int o = 16; o > 0; o >>= 1) v += __shfl_down(v, o, 32);
    if (lane == 0) dst[r] = v;
}

// ---------------------------------------------------------------------------
// Kernel 3: deterministic partial reduction of sum((yh-y)^2) and sum(relu(-yh))
// ---------------------------------------------------------------------------
__global__ __launch_bounds__(256) void reduce_kernel(const float* __restrict__ yh,
                                                     const float* __restrict__ y,
                                                     float* __restrict__ partials) {
    __shared__ float s1[256];
    __shared__ float s2[256];
    float sq = 0.0f, rl = 0.0f;
    for (size_t idx = (size_t)blockIdx.x * 256 + threadIdx.x; idx < TOTAL_ELEMS;
         idx += (size_t)1024 * 256) {
        float a = yh[idx], b = y[idx];
        float d = a - b;
        sq += d * d;
        rl += fmaxf(-a, 0.0f);
    }
    const int t = threadIdx.x;
    s1[t] = sq; s2[t] = rl;
    __syncthreads();
    for (int o = 128; o > 0; o >>= 1) {
        if (t < o) { s1[t] += s1[t + o]; s2[t] += s2[t + o]; }
        __syncthreads();
    }
    if (t == 0) { partials[2 * blockIdx.x] = s1[0]; partials[2 * blockIdx.x + 1] = s2[0]; }
}

// ---------------------------------------------------------------------------
// Kernel 4: LDS-staged, double-buffered WMMA fp32 GEMM.
//   Dd[(i+j)*N + i] = nA[i] + nB[j] - 2 * dot(Arow_i, Brow_j)   over K=512
// Block = 256 threads = 8 waves = 2 (i) x 4 (j) -> 32 x 256 macro-tile.
// Per K-block (KB=16): async-stage A(32xKB) and B(256xKB) panels into LDS
// (double buffered), then each wave runs 4 k-steps x 4 WMMAs from LDS.
// ---------------------------------------------------------------------------
#define GEMM_IT 32
#define GEMM_JT 256
#define GEMM_KB 16
#define GEMM_RS 20                     // LDS row stride in floats (pad, 16B-aligned)
#define GEMM_NKB (C_LEN / GEMM_KB)     // 32

__global__ __launch_bounds__(256) void gemm_kernel(const float* __restrict__ yh,
                                                   const float* __restrict__ y,
                                                   const float* __restrict__ normX,
                                                   const float* __restrict__ normY,
                                                   float* __restrict__ Dbuf,
                                                   int N, int Bwin, int step, int itemBase) {
    __shared__ float As[2][GEMM_IT * GEMM_RS];   // 2 x 2.5 KB
    __shared__ float Bs[2][GEMM_JT * GEMM_RS];   // 2 x 20 KB

    const int t    = threadIdx.x;
    const int wave = t >> 5;
    const int lane = t & 31;
    const int li   = lane & 15;
    const int kl   = (lane >> 4) * 2;

    const int i00 = blockIdx.y * GEMM_IT;
    const int j00 = blockIdx.x * GEMM_JT;

    const int g    = itemBase + blockIdx.z;
    const int pair = g / Bwin;
    const int m    = g - pair * Bwin;
    const int s    = m >> 2;
    const int b    = m & 3;
    const int start = s * step;

    const float* Asrc = (pair == 2) ? y : yh;
    const float* nA   = (pair == 2) ? normY : normX;
    const float* Bsrc = (pair == 1) ? yh : y;
    const float* nB   = (pair == 1) ? normX : normY;

    const size_t rowbase = (size_t)b * T_LEN + start;

    // ---- panel fill: <=5 async-copy instructions per thread -------------
    auto fill = [&](int buf, int kb) {
        const int kof = kb * GEMM_KB;
        const int c   = (t & 3) * 4;          // 4-float (16B) chunk within K-block
        if (t < 128) {                        // A: 32 rows x 4 chunks
            const int r = t >> 2;
            int grow = i00 + r; if (grow >= N) grow = N - 1;
            copy16(Asrc + (rowbase + grow) * (size_t)C_LEN + kof + c,
                   &As[buf][r * GEMM_RS + c]);
        }
#pragma unroll
        for (int u = 0; u < 4; ++u) {         // B: 256 rows x 4 chunks
            const int br = (t >> 2) + u * 64;
            int gcol = j00 + br; if (gcol >= N) gcol = N - 1;
            copy16(Bsrc + (rowbase + gcol) * (size_t)C_LEN + kof + c,
                   &Bs[buf][br * GEMM_RS + c]);
        }
    };

    const int arow  = (wave >> 2) * 16 + li;  // 0..31 within A panel
    const int brow0 = (wave & 3) * 64;        // base row within B panel

    const v8f vzero = {};
    v8f acc0 = vzero, acc1 = vzero, acc2 = vzero, acc3 = vzero;

    fill(0, 0);
    for (int kb = 0; kb < GEMM_NKB; ++kb) {
        const int buf = kb & 1;
        if (kb + 1 < GEMM_NKB) { fill(buf ^ 1, kb + 1); waitAsyncLE4(); }
        else                   { waitAsync0(); }
        __syncthreads();

#pragma unroll
        for (int ks = 0; ks < GEMM_KB; ks += 4) {
            const v2f a  = *(const v2f*)&As[buf][arow * GEMM_RS + ks + kl];
            const v2f b0 = *(const v2f*)&Bs[buf][(brow0 +  0 + li) * GEMM_RS + ks + kl];
            const v2f b1 = *(const v2f*)&Bs[buf][(brow0 + 16 + li) * GEMM_RS + ks + kl];
            const v2f b2 = *(const v2f*)&Bs[buf][(brow0 + 32 + li) * GEMM_RS + ks + kl];
            const v2f b3 = *(const v2f*)&Bs[buf][(brow0 + 48 + li) * GEMM_RS + ks + kl];
            acc0 = __builtin_amdgcn_wmma_f32_16x16x4_f32(false, a, false, b0, (short)0, acc0, false, false);
            acc1 = __builtin_amdgcn_wmma_f32_16x16x4_f32(false, a, false, b1, (short)0, acc1, false, false);
            acc2 = __builtin_amdgcn_wmma_f32_16x16x4_f32(false, a, false, b2, (short)0, acc2, false, false);
            acc3 = __builtin_amdgcn_wmma_f32_16x16x4_f32(false, a, false, b3, (short)0, acc3, false, false);
        }
        __syncthreads();
    }

    // ---- epilogue: D in DIAGONAL-MAJOR layout --------------------------
    // C/D layout: lanes 0-15: M=r, N=lane; lanes 16-31: M=r+8, N=lane-16
    const int i0 = i00 + (wave >> 2) * 16;
    const int j0 = j00 + (wave & 3) * 64;
    if (i0 >= N || j0 >= N) return;           // after all barriers

    float* Dd = Dbuf + (size_t)blockIdx.z * (size_t)(2 * N - 1) * N;
    const int rbase = (lane < 16) ? 0 : 8;
    v8f accq[4] = {acc0, acc1, acc2, acc3};
#pragma unroll
    for (int q = 0; q < 4; ++q) {
        const int col = j0 + q * 16 + li;
        if (col >= N) continue;
        const float nyv = nB[rowbase + col];
#pragma unroll
        for (int r = 0; r < 8; ++r) {
            const int row = i0 + rbase + r;
            if (row < N)
                Dd[(size_t)(row + col) * N + row] =
                    nA[rowbase + row] + nyv - 2.0f * accq[q][r];
        }
    }
}

// ---------------------------------------------------------------------------
// Kernel 5: soft-DTW anti-diagonal wavefront on diagonal-major D.
// ---------------------------------------------------------------------------
__device__ __forceinline__ float softmin3f(float a, float b, float c) {
    float mn = fminf(fminf(a, b), c);
    float sum = expf((mn - a) * GAMMA_INV) + expf((mn - b) * GAMMA_INV) +
                expf((mn - c) * GAMMA_INV);
    return mn - GAMMA * logf(sum);
}

__global__ __launch_bounds__(512) void dtw_kernel(const float* __restrict__ Dbuf,
                                                  float* __restrict__ out,
                                                  int N, int itemBase) {
    __shared__ float buf[3][400];
    const float* Dd = Dbuf + (size_t)blockIdx.x * (size_t)(2 * N - 1) * N;
    const int i = threadIdx.x;
    const int P = 2 * N - 1;

    auto validAt = [&](int p) -> bool {
        const int j = p - i;
        return (i < N) && (j >= 0) && (j < N);
    };

    float dval = validAt(0) ? Dd[i] : 0.0f;
    for (int p = 0; p < P; ++p) {
        const int j = p - i;
        const bool valid = validAt(p);
        float v = 0.0f;
        if (valid) {
            const float a  = (i >= 1) ? buf[(p + 2) % 3][i - 1] : BIGV;
            const float bb = (j >= 1) ? buf[(p + 2) % 3][i]     : BIGV;
            float c;
            if (i >= 1 && j >= 1)      c = buf[(p + 1) % 3][i - 1];
            else if (i == 0 && j == 0) c = 0.0f;
            else                       c = BIGV;
            v = dval + softmin3f(a, bb, c);
        }
        float dn = 0.0f;
        if (p + 1 < P && validAt(p + 1)) dn = Dd[(size_t)(p + 1) * N + i];
        if (p + 2 < P && i < N) __builtin_prefetch(&Dd[(size_t)(p + 2) * N + i], 0, 3);
        if (valid) buf[p % 3][i] = v;
        __syncthreads();
        dval = dn;
    }
    if (i == N - 1) out[itemBase + blockIdx.x] = buf[(P - 1) % 3][N - 1];
}

// ---------------------------------------------------------------------------
// Kernel 6: final deterministic combine -> scalar loss
// ---------------------------------------------------------------------------
__device__ float blockReduceSum(float v, float* s) {
    const int t = threadIdx.x;
    s[t] = v;
    __syncthreads();
    for (int o = 128; o > 0; o >>= 1) {
        if (t < o) s[t] += s[t + o];
        __syncthreads();
    }
    float r = s[0];
    __syncthreads();
    return r;
}

__global__ __launch_bounds__(256) void final_kernel(const float* __restrict__ partials,
                                                    const float* __restrict__ dtw200,
                                                    const float* __restrict__ dtw400,
                                                    float* __restrict__ out) {
    __shared__ float s[256];
    const int t = threadIdx.x;

    float psq = 0.0f, prl = 0.0f;
    for (int i = t; i < 1024; i += 256) { psq += partials[2 * i]; prl += partials[2 * i + 1]; }
    const float sumsq = blockReduceSum(psq, s);
    const float sumrl = blockReduceSum(prl, s);

    float c200 = 0.0f;
    for (int mIt = t; mIt < 152; mIt += 256)
        c200 += dtw200[mIt] - 0.5f * (dtw200[152 + mIt] + dtw200[304 + mIt]);
    const float sum200 = blockReduceSum(c200, s);

    float c400 = 0.0f;
    for (int mIt = t; mIt < 72; mIt += 256)
        c400 += dtw400[mIt] - 0.5f * (dtw400[72 + mIt] + dtw400[144 + mIt]);
    const float sum400 = blockReduceSum(c400, s);

    if (t == 0) {
        const float M = (float)TOTAL_ELEMS;
        float loss = 0.1f * sqrtf(sumsq / M) + 0.5f * (sumrl / M);
        loss += 0.5f * 1e-5f * (sum200 / 152.0f);
        loss += 0.5f * 1e-5f * (sum400 / 72.0f);
        out[0] = loss;
    }
}

// ---------------------------------------------------------------------------
// Host launcher
// ---------------------------------------------------------------------------
extern "C" void kernel_launch(void* const* d_in, const int* in_sizes, int n_in,
                              void* d_out, int out_size, void* d_ws, size_t ws_size,
                              hipStream_t stream) {
    const float* y_hat = (const float*)d_in[0];   // (4, 512, 4000)
    const float* y     = (const float*)d_in[1];   // (4, 4000, 512)
    float* out = (float*)d_out;

    char* ws = (char*)d_ws;
    float* yh       = (float*)ws;                 // 8,192,000 floats
    float* normX    = yh + TOTAL_ELEMS;           // 16,000
    float* normY    = normX + NB * T_LEN;         // 16,000
    float* partials = normY + NB * T_LEN;         // 2048
    float* dtw200   = partials + 2048;            // 456
    float* dtw400   = dtw200 + 456;               // 216
    size_t fixedBytes = (size_t)((char*)(dtw400 + 216) - ws);
    size_t dOff = (fixedBytes + 255) & ~(size_t)255;
    float* Dbuf = (float*)(ws + dOff);
    size_t availD = (ws_size > dOff) ? (ws_size - dOff) : 0;

    {
        dim3 blk(32, 8, 1), grd(T_LEN / 32, C_LEN / 32, NB);
        transpose_kernel<<<grd, blk, 0, stream>>>(y_hat, yh);
    }
    norms_kernel<<<(2 * NB * T_LEN) / 8, 256, 0, stream>>>(yh, y, normX, normY);
    reduce_kernel<<<1024, 256, 0, stream>>>(yh, y, partials);

    struct Cfg { int N; int S; float* vals; };
    const Cfg cfg[2] = { {200, 38, dtw200}, {400, 18, dtw400} };
    for (int c = 0; c < 2; ++c) {
        const int N = cfg[c].N;
        const int Bwin = cfg[c].S * NB;       // 152 or 72
        const int total = 3 * Bwin;           // 456 or 216
        const int step = N / 2;
        const size_t bytesPer = (size_t)(2 * N - 1) * N * sizeof(float);  // diag-major
        int cap = availD ? (int)(availD / bytesPer) : 1;
        if (cap < 1) cap = 1;
        if (cap > total) cap = total;
        const int gx = (N + 255) / 256;
        const int gy = (N + 31) / 32;
        const int bdim = ((N + 31) / 32) * 32;
        for (int ofs = 0; ofs < total; ofs += cap) {
            const int cnt = (total - ofs < cap) ? (total - ofs) : cap;
            dim3 gg(gx, gy, cnt);
            gemm_kernel<<<gg, 256, 0, stream>>>(yh, y, normX, normY, Dbuf,
                                                N, Bwin, step, ofs);
            dtw_kernel<<<cnt, bdim, 0, stream>>>(Dbuf, cfg[c].vals, N, ofs);
        }
    }

    final_kernel<<<1, 256, 0, stream>>>(partials, dtw200, dtw400, out);
}